// TrajectoryConsistencyLoss_39058432590181
// MI455X (gfx1250) — compile-verified
//
#include <hip/hip_runtime.h>
#include <stdint.h>

// Trajectory consistency loss for positions [B=8192, S=2048, 2] fp32.
// Memory-bound: 134 MB -> 1 scalar (~5.8 us floor at 23.3 TB/s). Strategy:
//   Kernel 1: one workgroup per row; TDM tensor_load_to_lds pulls the 16 KB row
//             into LDS, threads compute second-difference norms + direct
//             5-wide sliding-window variances, block-reduce to 2 partials.
//   Kernel 2: single wave reduces all partials with v_wmma_f32_16x16x4_f32
//             (ones x data trick: permutation-invariant exact f32 sum).

#define SEQ   2048
#define WIN   5
#define NWIN  (SEQ - WIN + 1)   // 2044
#define NACC  (SEQ - 2)         // 2046

typedef float        v2f __attribute__((ext_vector_type(2)));
typedef float        v8f __attribute__((ext_vector_type(8)));
typedef unsigned int v4u __attribute__((ext_vector_type(4)));
typedef int          v4i __attribute__((ext_vector_type(4)));
typedef int          v8i __attribute__((ext_vector_type(8)));

__global__ __launch_bounds__(256) void traj_row_kernel(
    const float* __restrict__ pos,
    float* __restrict__ partial_s,
    float* __restrict__ partial_v) {
  __shared__ float2 row[SEQ];          // 16 KB
  __shared__ float red_s[8];
  __shared__ float red_v[8];

  const int tid = threadIdx.x;
  const uint64_t gaddr =
      (uint64_t)(uintptr_t)(pos + (size_t)blockIdx.x * (2u * SEQ));

  // ---- TDM: DMA this row (contiguous 16 KB) from global into LDS ----
  if (tid < 32) {  // wave 0 issues the descriptor (EXEC ignored by TDM)
    const uint32_t lds_off = (uint32_t)(uintptr_t)(&row[0]);  // low 32b = LDS offset
    v4u g0;
    g0.x = 1u;                                   // count=1, user mode
    g0.y = lds_off;                              // lds_addr
    g0.z = (uint32_t)gaddr;                      // global_addr[31:0]
    g0.w = ((uint32_t)(gaddr >> 32) & 0x01FFFFFFu) | 0x80000000u;  // addr[56:32] | type=2
    v8i g1;
    g1[0] = (int)(2u << 16);                     // wg_mask=0, data_size=2 (4 bytes)
    g1[1] = (int)((2u * SEQ) << 16);             // tensor_dim0 = 4096 (lo16)
    g1[2] = (int)(1u << 16);                     // tensor_dim0 hi16=0, tensor_dim1=1
    g1[3] = (int)((2u * SEQ) << 16);             // tile_dim0 = 4096
    g1[4] = 1;                                   // tile_dim1=1, tile_dim2=0
    g1[5] = (int)(2u * SEQ);                     // tensor_dim0_stride = 4096
    g1[6] = 0;
    g1[7] = 0;
    v4i gz4 = {0, 0, 0, 0};                      // groups 2/3 unused (<=2D tensor)
    v8i gz8 = {0, 0, 0, 0, 0, 0, 0, 0};          // clang-23 6-arg form: extra group
    __builtin_amdgcn_tensor_load_to_lds(g0, g1, gz4, gz4, gz8, 0);
    __builtin_amdgcn_s_wait_tensorcnt(0);
  }
  __syncthreads();

  // ---- per-thread contributions (strided: lane-consecutive LDS reads) ----
  float s_part = 0.0f;
  float v_part = 0.0f;
#pragma unroll
  for (int rep = 0; rep < SEQ / 256; ++rep) {
    const int i = tid + rep * 256;
    if (i < NACC) {
      const float2 p0 = row[i];
      const float2 p1 = row[i + 1];
      const float2 p2 = row[i + 2];
      const float ax = p2.x - 2.0f * p1.x + p0.x;
      const float ay = p2.y - 2.0f * p1.y + p0.y;
      s_part += sqrtf(ax * ax + ay * ay);
      if (i < NWIN) {
        const float2 p3 = row[i + 3];
        const float2 p4 = row[i + 4];
        const float sx = p0.x + p1.x + p2.x + p3.x + p4.x;
        const float sy = p0.y + p1.y + p2.y + p3.y + p4.y;
        const float qx = p0.x * p0.x + p1.x * p1.x + p2.x * p2.x +
                         p3.x * p3.x + p4.x * p4.x;
        const float qy = p0.y * p0.y + p1.y * p1.y + p2.y * p2.y +
                         p3.y * p3.y + p4.y * p4.y;
        // unbiased var: (sumsq - sum^2/5)/4, summed over x and y
        v_part += (qx - sx * sx * 0.2f) * 0.25f +
                  (qy - sy * sy * 0.2f) * 0.25f;
      }
    }
  }

  // ---- wave32 reduce, then cross-wave via LDS ----
#pragma unroll
  for (int off = 16; off > 0; off >>= 1) {
    s_part += __shfl_xor(s_part, off, 32);
    v_part += __shfl_xor(v_part, off, 32);
  }
  if ((tid & 31) == 0) {
    red_s[tid >> 5] = s_part;
    red_v[tid >> 5] = v_part;
  }
  __syncthreads();
  if (tid == 0) {
    float ss = 0.0f, vv = 0.0f;
#pragma unroll
    for (int w = 0; w < 8; ++w) { ss += red_s[w]; vv += red_v[w]; }
    partial_s[blockIdx.x] = ss;
    partial_v[blockIdx.x] = vv;
  }
}

// Single-wave WMMA reduction of per-row partials.
// D = Ones(16x4) x B(4x16) + C  => each D[m][n] = column-sum of B.
// Summing c[0] over all 32 lanes double-counts (rows 0 and 8 identical),
// so total = wavesum(c[0]) / 2. Exact fp32, permutation-invariant.
__global__ __launch_bounds__(32) void traj_reduce_kernel(
    const float* __restrict__ ps,
    const float* __restrict__ pv,
    float* __restrict__ out, int n) {
  const int lane = threadIdx.x;
  const v2f ones = {1.0f, 1.0f};
  v8f cs = {};
  v8f cv = {};
  const int full = n & ~63;
  for (int base = 0; base < full; base += 64) {
    v2f bs, bv;
    bs.x = ps[base + lane];
    bs.y = ps[base + 32 + lane];
    bv.x = pv[base + lane];
    bv.y = pv[base + 32 + lane];
    cs = __builtin_amdgcn_wmma_f32_16x16x4_f32(false, ones, false, bs,
                                               (short)0, cs, false, false);
    cv = __builtin_amdgcn_wmma_f32_16x16x4_f32(false, ones, false, bv,
                                               (short)0, cv, false, false);
  }
  float s = cs[0];
  float v = cv[0];
#pragma unroll
  for (int off = 16; off > 0; off >>= 1) {
    s += __shfl_xor(s, off, 32);
    v += __shfl_xor(v, off, 32);
  }
  s *= 0.5f;
  v *= 0.5f;
  if (lane == 0) {
    for (int i = full; i < n; ++i) { s += ps[i]; v += pv[i]; }  // tail (empty when n%64==0)
    const float inv_s = 1.0f / ((float)n * (float)NACC);
    const float inv_v = 1.0f / ((float)n * (float)NWIN * 2.0f);
    out[0] = 0.05f * s * inv_s + v * inv_v;
  }
}

extern "C" void kernel_launch(void* const* d_in, const int* in_sizes, int n_in,
                              void* d_out, int out_size, void* d_ws, size_t ws_size,
                              hipStream_t stream) {
  (void)n_in; (void)out_size; (void)ws_size;
  const float* pos = (const float*)d_in[0];
  const int nrows = in_sizes[0] / (2 * SEQ);   // 8192
  float* partial_s = (float*)d_ws;             // [nrows]
  float* partial_v = partial_s + nrows;        // [nrows]
  traj_row_kernel<<<nrows, 256, 0, stream>>>(pos, partial_s, partial_v);
  traj_reduce_kernel<<<1, 32, 0, stream>>>(partial_s, partial_v, (float*)d_out, nrows);
}